// KWinners2d_83983790506087
// MI455X (gfx1250) — compile-verified
//
#include <hip/hip_runtime.h>

#ifndef __has_builtin
#define __has_builtin(x) 0
#endif

#if __has_builtin(__builtin_amdgcn_global_load_async_to_lds_b128) && \
    __has_builtin(__builtin_amdgcn_s_wait_asynccnt)
#define HAVE_ASYNC 1
#else
#define HAVE_ASYNC 0
#endif

#define NT     1024                 // threads per block = 32 wave32 waves
#define NCH    256
#define NELEM  (NCH * 1024)         // 262144 elements per batch row
#define KSEL   26214u               // round(0.1 * 262144)
#define ITERS  (NELEM / 4 / NT)     // 64 float4-iterations per pass
#define NBUF   3                    // async staging buffers (depth-2 pipeline)

// Monotonic unsigned key: larger key <=> larger float (for non-NaN).
__device__ __forceinline__ unsigned fkey(float f) {
  unsigned u = __float_as_uint(f);
  return u ^ (((unsigned)((int)u >> 31)) | 0x80000000u);
}

#if HAVE_ASYNC
typedef int v4i __attribute__((__vector_size__(4 * sizeof(int))));
typedef __attribute__((address_space(1))) v4i* gptr_v4i;  // global
typedef __attribute__((address_space(3))) v4i* lptr_v4i;  // LDS

__device__ __forceinline__ void async_load16(const void* g, void* l) {
  __builtin_amdgcn_global_load_async_to_lds_b128(
      (gptr_v4i)g, (lptr_v4i)l, /*offset=*/0, /*cpol=*/0);
}
#endif

// Stream one batch row (as float4) through `f(i, v)`.
// USE_ASYNC: 3-buffer depth-2 global->LDS DMA pipeline (HBM pass). Each lane
// reads back only the LDS slot it staged itself, so the wave's own
// s_wait_asynccnt is the only synchronization needed.
// !USE_ASYNC: direct vector loads (rows are L2-resident after pass 0).
template <bool USE_ASYNC, class F>
__device__ __forceinline__ void stream_row(const float4* __restrict__ g,
                                           float4* lbuf, int t, F f) {
#if HAVE_ASYNC
  if constexpr (USE_ASYNC) {
    async_load16(g + t,      lbuf + t);
    async_load16(g + NT + t, lbuf + NT + t);
    int bi = 0;                               // buffer holding tile i
    for (int i = 0; i < ITERS; ++i) {
      if (i + 2 < ITERS) {
        int bn = bi + 2; if (bn >= NBUF) bn -= NBUF;
        async_load16(g + (i + 2) * NT + t, lbuf + bn * NT + t);
        __builtin_amdgcn_s_wait_asynccnt(2);  // in-order => tile i complete
      } else {
        __builtin_amdgcn_s_wait_asynccnt(0);  // drain tail (imm operand)
      }
      f(i, lbuf[bi * NT + t]);
      if (++bi == NBUF) bi = 0;
    }
    return;
  }
#endif
  for (int i = 0; i < ITERS; ++i) {
    if (i + 8 < ITERS) __builtin_prefetch(g + (i + 8) * NT + t, 0, 0);
    f(i, g[i * NT + t]);
  }
}

__global__ __launch_bounds__(NT)
void kwinners2d_kernel(const float* __restrict__ x,
                       const float* __restrict__ duty,
                       float* __restrict__ out) {
  __shared__ float    boostS[NCH];
  __shared__ unsigned hist[2048];
  __shared__ unsigned part[NT];
  __shared__ unsigned res[2];     // [0] = digit, [1] = count strictly above bucket
#if HAVE_ASYNC
  __shared__ float4   lbuf[NBUF * NT];        // 48 KB staging
#else
  float4* lbuf = nullptr;
#endif

  const int t = threadIdx.x;
  const float4* __restrict__ xb =
      (const float4*)(x + (size_t)blockIdx.x * NELEM);
  float4* __restrict__ ob = (float4*)(out + (size_t)blockIdx.x * NELEM);

  if (t < NCH) {
    const float td = (float)KSEL / (float)NELEM;   // target density
    boostS[t] = expf(td - duty[t]);                // BOOST_STRENGTH == 1
  }
  __syncthreads();

  unsigned kk = KSEL;          // rank remaining (uniform across threads)
  unsigned prefix = 0;         // resolved high bits of the threshold key
  unsigned thresholdKey = 0;

  // ---- 3-pass radix select on key bits [31:21], [20:10], [9:0] ----
  for (int pass = 0; pass < 3; ++pass) {
    const int nbins = (pass == 2) ? 1024 : 2048;
    for (int i = t; i < nbins; i += NT) hist[i] = 0;
    __syncthreads();

    auto consume = [&](int i, float4 v) {
      const int c = (i * NT + t) >> 8;             // elem_base>>10, base=(i*NT+t)*4
      const float bst = boostS[c];
      unsigned k0 = fkey(v.x * bst), k1 = fkey(v.y * bst);
      unsigned k2 = fkey(v.z * bst), k3 = fkey(v.w * bst);
      if (pass == 0) {
        atomicAdd(&hist[k0 >> 21], 1u); atomicAdd(&hist[k1 >> 21], 1u);
        atomicAdd(&hist[k2 >> 21], 1u); atomicAdd(&hist[k3 >> 21], 1u);
      } else if (pass == 1) {
        if ((k0 >> 21) == prefix) atomicAdd(&hist[(k0 >> 10) & 0x7FFu], 1u);
        if ((k1 >> 21) == prefix) atomicAdd(&hist[(k1 >> 10) & 0x7FFu], 1u);
        if ((k2 >> 21) == prefix) atomicAdd(&hist[(k2 >> 10) & 0x7FFu], 1u);
        if ((k3 >> 21) == prefix) atomicAdd(&hist[(k3 >> 10) & 0x7FFu], 1u);
      } else {
        if ((k0 >> 10) == prefix) atomicAdd(&hist[k0 & 0x3FFu], 1u);
        if ((k1 >> 10) == prefix) atomicAdd(&hist[k1 & 0x3FFu], 1u);
        if ((k2 >> 10) == prefix) atomicAdd(&hist[k2 & 0x3FFu], 1u);
        if ((k3 >> 10) == prefix) atomicAdd(&hist[k3 & 0x3FFu], 1u);
      }
    };
    if (pass == 0) stream_row<true >(xb, lbuf, t, consume);  // HBM: async DMA
    else           stream_row<false>(xb, lbuf, t, consume);  // L2-resident
    __syncthreads();

    // Block-wide suffix sums over the histogram, then locate the bucket
    // containing rank kk (counting from the largest bin downwards).
    const int ch = nbins / NT;            // 2 or 1 bins per thread
    unsigned s = 0;
    for (int j = 0; j < ch; ++j) s += hist[t * ch + j];
    part[t] = s;
    __syncthreads();
    for (int off = 1; off < NT; off <<= 1) {       // inclusive suffix scan
      unsigned v = part[t] + ((t + off < NT) ? part[t + off] : 0u);
      __syncthreads();
      part[t] = v;
      __syncthreads();
    }
    unsigned run = (t == NT - 1) ? 0u : part[t + 1];  // S(chunk_end)
    for (int j = ch - 1; j >= 0; --j) {
      unsigned h  = hist[t * ch + j];
      unsigned sj = run + h;                          // S(bin j)
      if (sj >= kk && run < kk) {                     // unique crossing bin
        res[0] = (unsigned)(t * ch + j);
        res[1] = run;
      }
      run = sj;
    }
    __syncthreads();
    const unsigned d = res[0];
    kk -= res[1];
    if (pass == 0)      prefix = d;
    else if (pass == 1) prefix = (prefix << 11) | d;
    else                thresholdKey = (prefix << 10) | d;
    __syncthreads();   // protect hist/part/res before next pass reuses them
  }

  // ---- mask pass: out = (key(x*boost) >= T) ? x : 0  (ties included) ----
  stream_row<false>(xb, lbuf, t, [&](int i, float4 v) {
    const int c = (i * NT + t) >> 8;
    const float bst = boostS[c];
    float4 o;
    o.x = (fkey(v.x * bst) >= thresholdKey) ? v.x : 0.0f;
    o.y = (fkey(v.y * bst) >= thresholdKey) ? v.y : 0.0f;
    o.z = (fkey(v.z * bst) >= thresholdKey) ? v.z : 0.0f;
    o.w = (fkey(v.w * bst) >= thresholdKey) ? v.w : 0.0f;
    ob[i * NT + t] = o;
  });
}

extern "C" void kernel_launch(void* const* d_in, const int* in_sizes, int n_in,
                              void* d_out, int out_size, void* d_ws, size_t ws_size,
                              hipStream_t stream) {
  const float* x    = (const float*)d_in[0];
  const float* duty = (const float*)d_in[1];
  float* out        = (float*)d_out;
  const int batches = in_sizes[0] / NELEM;   // 128
  kwinners2d_kernel<<<dim3(batches), dim3(NT), 0, stream>>>(x, duty, out);
}